// DevConv_35364760715802
// MI455X (gfx1250) — compile-verified
//
#include <hip/hip_runtime.h>

// N = 8192 nodes, 16x16 distance tiles, gram tile via V_WMMA_F32_16X16X4_F32.
#define NN 8192
#define TILES (NN / 16)       // 512 column tiles per row-slab
#define WAVES_PER_BLOCK 8     // 256 threads, wave32

typedef float v2f __attribute__((ext_vector_type(2)));
typedef float v8f __attribute__((ext_vector_type(8)));

__global__ __launch_bounds__(256) void devconv_maxdist_kernel(
    const float* __restrict__ prev,     // [N]
    const float* __restrict__ nodes,    // [N,3]
    const int*   __restrict__ adj,      // [N,N]
    const float* __restrict__ Wphi,     // [32]
    const float* __restrict__ Wth,      // [3]
    float* __restrict__ out)            // [N]
{
    __shared__ float red[WAVES_PER_BLOCK * 16];

    const int tid  = threadIdx.x;
    const int lane = tid & 31;
    // wave id is uniform across the wave: force it into an SGPR so the tile
    // loop + prefetch guard compile to scalar branches (EXEC untouched, as
    // WMMA requires all-1s EXEC).
    const int wv   = __builtin_amdgcn_readfirstlane(tid >> 5);   // 0..7
    const int half = lane >> 4;         // 0 = lanes 0-15, 1 = lanes 16-31
    const int l15  = lane & 15;
    const int rowBase = blockIdx.x * 16;

    // uniform weights
    const float w0 = Wth[0], w1 = Wth[1], w2 = Wth[2];

    // ---- A fragment (16x4, K padded to 4 with zeros), fixed per block ----
    // lanes 0-15: (y.x, y.y) of row M=l15 ; lanes 16-31: (y.z, 0)
    const int m = rowBase + l15;
    const float ax = nodes[3 * m + 0] * w0;
    const float ay = nodes[3 * m + 1] * w1;
    const float az = nodes[3 * m + 2] * w2;
    const float sq_m = ax * ax + ay * ay + az * az;
    v2f afrag;
    afrag.x = half ? az : ax;
    afrag.y = half ? 0.0f : ay;

    // sqrow[v] = |y_row|^2 for row M = rowBase + v + 8*half (C-matrix layout).
    // That value was computed by lane (v + 8*half); broadcast via shuffle.
    float sqrow[8];
#pragma unroll
    for (int v = 0; v < 8; ++v)
        sqrow[v] = __shfl(sq_m, v + 8 * half, 32);

    // acc tracks max_j (sq_n - 2*gram); the uniform-per-row sq_m term is
    // added after the loop (adding to -inf keeps -inf for isolated rows).
    float acc[8];
#pragma unroll
    for (int v = 0; v < 8; ++v) acc[v] = -__builtin_inff();

    // adjacency base for this lane's half: row (rowBase + 8*half)
    const long adjRowBase = (long)(rowBase + 8 * half) * NN;

    for (int t = wv; t < TILES; t += WAVES_PER_BLOCK) {
        const int colBase = t * 16;
        const int n = colBase + l15;

        // prefetch this wave's next adjacency tile (WGP scope -> all levels)
        if (t + WAVES_PER_BLOCK < TILES)
            __builtin_prefetch(adj + adjRowBase + n + 16 * WAVES_PER_BLOCK, 0, 3);

        // ---- B fragment (4x16): same (x,y | z,0) half packing as A ----
        const float bx = nodes[3 * n + 0] * w0;
        const float by = nodes[3 * n + 1] * w1;
        const float bz = nodes[3 * n + 2] * w2;
        const float sq_n = bx * bx + by * by + bz * bz;
        v2f bfrag;
        bfrag.x = half ? bz : bx;
        bfrag.y = half ? 0.0f : by;

        // gram tile = A x B  (16x16 f32 in 8 VGPRs)
        v8f c = {};
        c = __builtin_amdgcn_wmma_f32_16x16x4_f32(
                /*neg_a=*/false, afrag, /*neg_b=*/false, bfrag,
                /*c_mod=*/(short)0, c, /*reuse_a=*/false, /*reuse_b=*/false);

        // masked running max of (sq_n - 2*gram)
#pragma unroll
        for (int v = 0; v < 8; ++v) {
            const int a = adj[adjRowBase + (long)v * NN + n];   // coalesced b32
            float d = fmaf(-2.0f, c[v], sq_n);
            d = (a > 0) ? d : -__builtin_inff();
            acc[v] = fmaxf(acc[v], d);
        }
    }

    // ---- reduce across N within each 16-lane half (xor shuffles) ----
#pragma unroll
    for (int v = 0; v < 8; ++v) {
        float mx = acc[v] + sqrow[v];     // restore the sq_m term (-inf safe)
        mx = fmaxf(mx, __shfl_xor(mx, 8, 32));
        mx = fmaxf(mx, __shfl_xor(mx, 4, 32));
        mx = fmaxf(mx, __shfl_xor(mx, 2, 32));
        mx = fmaxf(mx, __shfl_xor(mx, 1, 32));
        if (l15 == 0) red[wv * 16 + v + 8 * half] = mx;  // lanes 0 and 16 write
    }
    __syncthreads();

    // ---- combine 8 waves, finish scalar math, write 16 outputs ----
    if (tid < 16) {
        float mx = red[tid];
#pragma unroll
        for (int w = 1; w < WAVES_PER_BLOCK; ++w)
            mx = fmaxf(mx, red[w * 16 + tid]);
        const float maxdist = sqrtf(fmaxf(mx, 0.0f));

        float s = 0.0f;
#pragma unroll
        for (int k = 0; k < 32; ++k) s += Wphi[k];
        const float meanphi = s * (1.0f / 32.0f);

        const int row = rowBase + tid;
        out[row] = (prev[row] + maxdist * meanphi) * 0.5f;
    }
}

extern "C" void kernel_launch(void* const* d_in, const int* in_sizes, int n_in,
                              void* d_out, int out_size, void* d_ws, size_t ws_size,
                              hipStream_t stream) {
    const float* prev  = (const float*)d_in[0];  // previous_inclusion_score [8192]
    const float* nodes = (const float*)d_in[1];  // nodes [8192,3]
    const int*   adj   = (const int*)d_in[2];    // adjacency_matrix [8192,8192]
    const float* wphi  = (const float*)d_in[3];  // W_phi [32]
    const float* wth   = (const float*)d_in[4];  // W_theta [3,1]
    float* out = (float*)d_out;

    devconv_maxdist_kernel<<<NN / 16, 256, 0, stream>>>(prev, nodes, adj, wphi, wth, out);
}